// Attention_org_10514079941402
// MI455X (gfx1250) — compile-verified
//
#include <hip/hip_runtime.h>

typedef float v2f __attribute__((ext_vector_type(2)));
typedef float v8f __attribute__((ext_vector_type(8)));

#define Tn 8
#define Dn 2048
#define Hn 4
#define EPSF 1e-5f

// LDS layout (floats)
#define SE_SZ   (16 * Dn)   // stacked emb pair: rows 0-7 = n0, 8-15 = n1
#define SRED_SZ (8 * 256)   // per-wave partial 16x16 Gram tiles
#define SMEM_FLOATS (SE_SZ + SRED_SZ + 256 /*sG*/ + 256*3 /*Wq,Wk,Wv*/ + 64 /*Wo*/ + 64 /*rel*/ \
                     + 128 /*sSc*/ + 128 /*sNrm*/ + 128 /*sP*/ + 128 /*sM*/ + 256 /*sA16*/)

__global__ __launch_bounds__(256) void attn_gram_fused(
    const float* __restrict__ emb, const float* __restrict__ Wq,
    const float* __restrict__ Wk,  const float* __restrict__ Wv,
    const float* __restrict__ Wo,  const float* __restrict__ rel,
    float* __restrict__ out)
{
    extern __shared__ float smem[];
    float* sE   = smem;               // 16 x 2048
    float* sRed = sE   + SE_SZ;       // 8 x 256
    float* sG   = sRed + SRED_SZ;     // 16x16 stacked Gram
    float* sWq  = sG   + 256;
    float* sWk  = sWq  + 256;
    float* sWv  = sWk  + 256;
    float* sWo  = sWv  + 256;
    float* sRel = sWo  + 64;
    float* sSc  = sRel + 64;          // 2 x 64 scores / exp scratch
    float* sNrm = sSc  + 128;         // 2 x 64 normalized scores
    float* sP   = sNrm + 128;         // 2 x 64 probs
    float* sM   = sP   + 128;         // 2 x 64 M matrices
    float* sA16 = sM   + 128;         // block-diag A, 16x16

    const int tid  = threadIdx.x;
    const int lane = tid & 31;
    const int wave = tid >> 5;
    const int n0   = blockIdx.x * 2;

    // ---- Stage emb pair into LDS: 16x2048 contiguous in memory -> contiguous in LDS
    {
        const float4* src4 = (const float4*)(emb + (size_t)n0 * Tn * Dn);
        float4* dst4 = (float4*)sE;
        for (int i = tid; i < (16 * Dn) / 4; i += 256) dst4[i] = src4[i];
    }
    // ---- Stage tiny weights
    sWq[tid] = Wq[tid];  sWk[tid] = Wk[tid];  sWv[tid] = Wv[tid];
    if (tid < 64) sWo[tid] = Wo[tid];
    if (tid < (2 * Tn - 1) * Hn) sRel[tid] = rel[tid];
    __syncthreads();

    // ================= Phase A: Gram of stacked pair via WMMA f32 16x16x4 =================
    // C16[r][c] = sum_d sE[r][d]*sE[c][d]; diag 8x8 blocks are G[n0], G[n1].
    // 32-bit A layout: lane<16 -> {k0,k1}, lane>=16 -> {k2,k3}; B fetch is identical by symmetry.
    const int mrow = lane & 15;
    const int kb   = (lane >> 4) << 1;
    v8f cg = {0.f, 0.f, 0.f, 0.f, 0.f, 0.f, 0.f, 0.f};
    {
        const float* rowPtr = sE + mrow * Dn;
        const int k0 = wave * (Dn / 8);
        for (int k = k0; k < k0 + (Dn / 8); k += 4) {
            v2f a = *(const v2f*)(rowPtr + k + kb);  // == matching B operand (Gram symmetry)
            cg = __builtin_amdgcn_wmma_f32_16x16x4_f32(false, a, false, a, (short)0, cg,
                                                       false, false);
        }
        float* red = sRed + wave * 256;
#pragma unroll
        for (int v = 0; v < 8; ++v) {
            int M = v + ((lane >> 4) << 3);            // C/D layout: VGPR v, lane-half -> row
            red[M * 16 + (lane & 15)] = cg[v];
        }
    }
    __syncthreads();
    {   // cross-wave reduction of the 16x16 tile
        float s = 0.f;
#pragma unroll
        for (int w = 0; w < 8; ++w) s += sRed[w * 256 + tid];
        sG[tid] = s;
    }
    __syncthreads();

    // ================= Phase B: per-(n,h) 8x8 algebra (threads 0..127 active) ============
    const int g  = tid >> 6;          // 0,1 = sample slot; 2,3 idle
    const int l  = tid & 63;
    const int tt = l >> 3, ss = l & 7;
    float Macc = 0.f;
    float myval = 0.f;

    for (int h = 0; h < Hn; ++h) {
        if (g < 2) {  // scores = Wq G Wk^T / sqrt(T) + bias
            const float* G  = sG + (g * 8) * 16 + g * 8;   // G[i][j] at G[i*16+j]
            const float* wq = sWq + h * 64 + tt * 8;
            const float* wk = sWk + h * 64 + ss * 8;
            float acc = 0.f;
#pragma unroll
            for (int j = 0; j < 8; ++j) {
                float qt = 0.f;
#pragma unroll
                for (int i = 0; i < 8; ++i) qt += wq[i] * G[i * 16 + j];
                acc += qt * wk[j];
            }
            myval = acc * 0.35355339059327373f + sRel[(tt - ss + Tn - 1) * Hn + h];
            sSc[g * 64 + l] = myval;
        }
        __syncthreads();
        if (g < 2) {  // instance norm over the 64-element (T,T) map
            float mu = 0.f, m2 = 0.f;
#pragma unroll 8
            for (int i = 0; i < 64; ++i) { float v = sSc[g * 64 + i]; mu += v; m2 += v * v; }
            mu *= (1.f / 64.f); m2 *= (1.f / 64.f);
            myval = (myval - mu) * rsqrtf((m2 - mu * mu) + EPSF);
            sNrm[g * 64 + l] = myval;
        }
        __syncthreads();
        if (g < 2) {  // softmax over key axis: row max + exp
            const float* row = sNrm + g * 64 + tt * 8;
            float mx = row[0];
#pragma unroll
            for (int j = 1; j < 8; ++j) mx = fmaxf(mx, row[j]);
            sSc[g * 64 + l] = __expf(myval - mx);
        }
        __syncthreads();
        if (g < 2) {  // normalize + store probs
            const float* row = sSc + g * 64 + tt * 8;
            float rs = 0.f;
#pragma unroll
            for (int j = 0; j < 8; ++j) rs += row[j];
            sP[g * 64 + l] = sSc[g * 64 + l] / rs;
        }
        __syncthreads();
        if (g < 2) {  // M += (1/H) * probs @ Wv[h]
            const float* prow = sP + g * 64 + tt * 8;
            const float* wv   = sWv + h * 64;
            float m = 0.f;
#pragma unroll
            for (int j = 0; j < 8; ++j) m += prow[j] * wv[j * 8 + ss];
            Macc += 0.25f * m;
        }
        __syncthreads();
    }

    if (g < 2) sM[g * 64 + l] = Macc;
    sA16[tid] = 0.f;                       // zero block-diagonal A
    __syncthreads();
    if (g < 2) {  // A = Wo @ M, placed block-diagonally into 16x16
        const float* mg = sM + g * 64;
        const float* wo = sWo + tt * 8;
        float a = 0.f;
#pragma unroll
        for (int j = 0; j < 8; ++j) a += wo[j] * mg[j * 8 + ss];
        sA16[(g * 8 + tt) * 16 + (g * 8 + ss)] = a;
    }
    __syncthreads();

    // ================= Phase C: out_pair = diag(A0,A1) @ emb_pair via WMMA ================
    v2f aOp[4];
#pragma unroll
    for (int j = 0; j < 4; ++j) {          // A operand chunks, K = 4j..4j+3
        aOp[j].x = sA16[mrow * 16 + 4 * j + kb];
        aOp[j].y = sA16[mrow * 16 + 4 * j + kb + 1];
    }
    float* outBase = out + (size_t)n0 * Tn * Dn;   // 16 contiguous rows of 2048
    for (int dt = wave; dt < Dn / 16; dt += 8) {
        const int d0 = dt * 16;
        v8f co = {0.f, 0.f, 0.f, 0.f, 0.f, 0.f, 0.f, 0.f};
#pragma unroll
        for (int j = 0; j < 4; ++j) {      // B[k'][c] = sE[4j+k'][d0+c]
            v2f b;
            b.x = sE[(4 * j + kb) * Dn + d0 + (lane & 15)];
            b.y = sE[(4 * j + kb + 1) * Dn + d0 + (lane & 15)];
            co = __builtin_amdgcn_wmma_f32_16x16x4_f32(false, aOp[j], false, b, (short)0, co,
                                                       false, false);
        }
#pragma unroll
        for (int v = 0; v < 8; ++v) {
            int M = v + ((lane >> 4) << 3);
            outBase[(size_t)M * Dn + d0 + (lane & 15)] = co[v];
        }
    }
}

extern "C" void kernel_launch(void* const* d_in, const int* in_sizes, int n_in,
                              void* d_out, int out_size, void* d_ws, size_t ws_size,
                              hipStream_t stream) {
    (void)n_in; (void)out_size; (void)d_ws; (void)ws_size;
    const float* emb = (const float*)d_in[0];
    const float* Wq  = (const float*)d_in[1];
    const float* Wk  = (const float*)d_in[2];
    const float* Wv  = (const float*)d_in[3];
    const float* Wo  = (const float*)d_in[4];
    const float* rel = (const float*)d_in[5];
    float* out = (float*)d_out;

    const int N = in_sizes[0] / (Tn * Dn);       // 2048
    const size_t shmem = SMEM_FLOATS * sizeof(float);   // ~143.5 KB, fits 320 KB/WGP
    (void)hipFuncSetAttribute((const void*)attn_gram_fused,
                              hipFuncAttributeMaxDynamicSharedMemorySize, (int)shmem);
    attn_gram_fused<<<dim3(N / 2), dim3(256), shmem, stream>>>(emb, Wq, Wk, Wv, Wo, rel, out);
}